// EI_Adaptive_Fusion_72653666779472
// MI455X (gfx1250) — compile-verified
//
#include <hip/hip_runtime.h>
#include <math.h>

#define PI_F 3.14159265358979323846f

typedef float v2f __attribute__((ext_vector_type(2)));
typedef float v8f __attribute__((ext_vector_type(8)));

// ---------------------------------------------------------------------------
// CDNA5 async global->LDS copy (b128), with portable fallback.
// Builtin prototype (probe-confirmed via round-2 diagnostic):
//   void __builtin_amdgcn_global_load_async_to_lds_b128(
//        global v4i* src, local v4i* dst, imm int offset, imm int cpol)
// ---------------------------------------------------------------------------
#if defined(__has_builtin)
#if __has_builtin(__builtin_amdgcn_global_load_async_to_lds_b128)
#define USE_ASYNC_LDS 1
#endif
#endif

typedef int av4i __attribute__((__vector_size__(16)));
typedef __attribute__((address_space(1))) av4i* gav4i_p;
typedef __attribute__((address_space(3))) av4i* lav4i_p;

__device__ __forceinline__ void async_copy_b128(const float* g, float* l)
{
#ifdef USE_ASYNC_LDS
    gav4i_p gp = (gav4i_p)(uintptr_t)g;
    lav4i_p lp = (lav4i_p)(unsigned)(uintptr_t)l;   // LDS offset = low 32 bits
    __builtin_amdgcn_global_load_async_to_lds_b128(gp, lp, 0, 0);
#else
    *(float4*)l = *(const float4*)g;
#endif
}

__device__ __forceinline__ void wait_async_copies()
{
#ifdef USE_ASYNC_LDS
#if __has_builtin(__builtin_amdgcn_s_wait_asynccnt)
    __builtin_amdgcn_s_wait_asynccnt(0);
#else
    asm volatile("s_wait_asynccnt 0" ::: "memory");
#endif
#endif
}

// ---------------------------------------------------------------------------
// fp32 WMMA GEMM: C(16x64 per wave) = alpha * A @ B [+ bias[n]]
//   A: row-major, lda
//   opB==1: B = W (N x K row-major): B[k][n] = Bp[n*ldb + bcol + k]  (W^T)
//   opB==0: B row-major KxN:         B[k][n] = Bp[k*ldb + bcol + n]
// grid = (N/64, M/16), block = 32 (one wave). K multiple of 4, K >= 8.
// 2-deep register pipeline: fragments for step k+1 are loaded before the
// 4 WMMAs of step k issue, so loadcnt waits overlap matrix ops.
// Fragment layouts per ISA 7.12.2: A 16x4 f32 -> 2 VGPRs (lane m, K=2*half+j);
// C/D 16x16 f32 -> 8 VGPRs (VGPR r = rows r, r+8 across lane halves).
// ---------------------------------------------------------------------------
__global__ __launch_bounds__(32)
void wmma_gemm_f32(const float* __restrict__ A, const float* __restrict__ B,
                   float* __restrict__ C, const float* __restrict__ bias,
                   int K, int lda, int ldb, int ldc,
                   int bcol, int opB, float alpha)
{
    const int lane  = threadIdx.x;
    const int lhalf = lane >> 4;           // 0: K={0,1}, 1: K={2,3}
    const int l16   = lane & 15;
    const int m0    = blockIdx.y * 16;
    const int n0    = blockIdx.x * 64;

    v8f acc[4] = {};
    const float* Arow = A + (size_t)(m0 + l16) * lda;

    if (opB) {
        const float* Br[4];
#pragma unroll
        for (int j = 0; j < 4; ++j)
            Br[j] = B + (size_t)(n0 + 16 * j + l16) * ldb + bcol;

        v2f a_cur = *(const v2f*)(Arow + 2 * lhalf);
        v2f b_cur[4];
#pragma unroll
        for (int j = 0; j < 4; ++j)
            b_cur[j] = *(const v2f*)(Br[j] + 2 * lhalf);

        for (int k0 = 4; k0 < K; k0 += 4) {
            const int ka = k0 + 2 * lhalf;
            const v2f a_nxt = *(const v2f*)(Arow + ka);
            v2f b_nxt[4];
#pragma unroll
            for (int j = 0; j < 4; ++j)
                b_nxt[j] = *(const v2f*)(Br[j] + ka);
#pragma unroll
            for (int j = 0; j < 4; ++j)
                acc[j] = __builtin_amdgcn_wmma_f32_16x16x4_f32(
                             false, a_cur, false, b_cur[j], (short)0, acc[j],
                             false, false);
            a_cur = a_nxt;
#pragma unroll
            for (int j = 0; j < 4; ++j) b_cur[j] = b_nxt[j];
        }
#pragma unroll
        for (int j = 0; j < 4; ++j)
            acc[j] = __builtin_amdgcn_wmma_f32_16x16x4_f32(
                         false, a_cur, false, b_cur[j], (short)0, acc[j],
                         false, false);
    } else {
        const float* Bc = B + bcol + n0 + l16;

        v2f a_cur = *(const v2f*)(Arow + 2 * lhalf);
        v2f b_cur[4];
#pragma unroll
        for (int j = 0; j < 4; ++j) {
            b_cur[j][0] = Bc[(size_t)(2 * lhalf + 0) * ldb + 16 * j];
            b_cur[j][1] = Bc[(size_t)(2 * lhalf + 1) * ldb + 16 * j];
        }

        for (int k0 = 4; k0 < K; k0 += 4) {
            const int ka = k0 + 2 * lhalf;
            const v2f a_nxt = *(const v2f*)(Arow + ka);
            v2f b_nxt[4];
#pragma unroll
            for (int j = 0; j < 4; ++j) {
                b_nxt[j][0] = Bc[(size_t)(ka + 0) * ldb + 16 * j];
                b_nxt[j][1] = Bc[(size_t)(ka + 1) * ldb + 16 * j];
            }
#pragma unroll
            for (int j = 0; j < 4; ++j)
                acc[j] = __builtin_amdgcn_wmma_f32_16x16x4_f32(
                             false, a_cur, false, b_cur[j], (short)0, acc[j],
                             false, false);
            a_cur = a_nxt;
#pragma unroll
            for (int j = 0; j < 4; ++j) b_cur[j] = b_nxt[j];
        }
#pragma unroll
        for (int j = 0; j < 4; ++j)
            acc[j] = __builtin_amdgcn_wmma_f32_16x16x4_f32(
                         false, a_cur, false, b_cur[j], (short)0, acc[j],
                         false, false);
    }

#pragma unroll
    for (int j = 0; j < 4; ++j) {
        const int n    = n0 + 16 * j + l16;
        const float bv = bias ? bias[n] : 0.0f;
#pragma unroll
        for (int r = 0; r < 8; ++r) {
            const int m = m0 + r + 8 * lhalf;
            C[(size_t)m * ldc + n] = acc[j][r] * alpha + bv;
        }
    }
}

// ---------------------------------------------------------------------------
// Prep: q[n][c] = erp[c*512 + n]; ERP spherical coords qc[512][3];
//       wsum[t] = sum_r Wdelta[r][t]
// ---------------------------------------------------------------------------
__global__ __launch_bounds__(256)
void prep_kernel(const float* __restrict__ erp, const float* __restrict__ Wdelta,
                 float* __restrict__ q, float* __restrict__ qc,
                 float* __restrict__ wsum)
{
    const int idx = blockIdx.x * blockDim.x + threadIdx.x;
    if (idx < 512 * 128) {
        const int n = idx >> 7, c = idx & 127;
        q[idx] = erp[c * 512 + n];
    }
    if (idx < 512) {
        const int y = idx >> 5, x = idx & 31;           // H=16, W=32
        const float u = (x - 16.5f) * (2.0f * PI_F / 32.0f);
        const float v = (y - 8.5f) * (PI_F / 16.0f);
        const float cv = cosf(v), sv = sinf(v);
        const float su = sinf(u), cu = cosf(u);
        qc[idx * 3 + 0] = cv * su;
        qc[idx * 3 + 1] = sv;
        qc[idx * 3 + 2] = cv * cu;
    }
    if (idx < 3) {
        float s = 0.0f;
        for (int r = 0; r < 128; ++r) s += Wdelta[r * 3 + idx];
        wsum[idx] = s;
    }
}

// ---------------------------------------------------------------------------
// DA affinity (transcendental hot loop, ~84M v_exp_f32):
//   D[q][k] = ( sum_c exp(-|q_da[q][c]-k_da[k][c]|)
//             + sum_t wsum[t]*exp(-|qc[q][t]-kc[k][t]|) ) / 128
// Block = 512 threads = 16 q-rows x 32 k-cols tile.
// Tiles staged to LDS with CDNA5 async b128 copies (ASYNCcnt-tracked),
// +4-float row padding kills 64-bank conflicts for the stride-128 access.
// ---------------------------------------------------------------------------
__global__ __launch_bounds__(512)
void da_affinity_kernel(const float* __restrict__ qda, const float* __restrict__ kvda,
                        const float* __restrict__ qc,  const float* __restrict__ kc,
                        const float* __restrict__ wsum, float* __restrict__ D)
{
    __shared__ float qt[16][132];
    __shared__ float kt[32][132];
    __shared__ float qct[16][3];
    __shared__ float kct[32][3];
    __shared__ float ws[3];

    const int tid = threadIdx.x;
    const int qb  = blockIdx.y * 16;
    const int kb  = blockIdx.x * 32;

    // q tile: 16 rows x 32 b128-chunks  (512 chunks, one per thread)
    {
        const int row = tid >> 5, ch = tid & 31;
        async_copy_b128(qda + (size_t)(qb + row) * 128 + ch * 4, &qt[row][ch * 4]);
    }
    // k tile: 32 rows x 32 b128-chunks  (1024 chunks, two per thread)
#pragma unroll
    for (int i = tid; i < 32 * 32; i += 512) {
        const int row = i >> 5, ch = i & 31;
        async_copy_b128(kvda + (size_t)(kb + row) * 256 + ch * 4, &kt[row][ch * 4]);
    }
    if (tid < 48) qct[tid / 3][tid % 3] = qc[qb * 3 + tid];
    if (tid < 96) kct[tid / 3][tid % 3] = kc[kb * 3 + tid];
    if (tid < 3)  ws[tid] = wsum[tid];
    wait_async_copies();
    __syncthreads();

    const int ty = tid >> 5;   // query in tile
    const int tx = tid & 31;   // key in tile
    float s = 0.0f;
#pragma unroll 8
    for (int c = 0; c < 128; ++c)
        s += __expf(-fabsf(qt[ty][c] - kt[tx][c]));
    float p = 0.0f;
#pragma unroll
    for (int t = 0; t < 3; ++t)
        p += ws[t] * __expf(-fabsf(qct[ty][t] - kct[tx][t]));
    D[(size_t)(qb + ty) * 1280 + (kb + tx)] = (s + p) * (1.0f / 128.0f);
}

// ---------------------------------------------------------------------------
// In-place row softmax, one block per row.
// ---------------------------------------------------------------------------
__global__ __launch_bounds__(256)
void softmax_rows_kernel(float* __restrict__ S, int ncols)
{
    __shared__ float red[256];
    float* Sr = S + (size_t)blockIdx.x * ncols;
    const int tid = threadIdx.x;

    float m = -3.402823466e38f;
    for (int c = tid; c < ncols; c += 256) m = fmaxf(m, Sr[c]);
    red[tid] = m; __syncthreads();
    for (int s = 128; s > 0; s >>= 1) {
        if (tid < s) red[tid] = fmaxf(red[tid], red[tid + s]);
        __syncthreads();
    }
    m = red[0]; __syncthreads();

    float sum = 0.0f;
    for (int c = tid; c < ncols; c += 256) {
        const float e = __expf(Sr[c] - m);
        Sr[c] = e;
        sum += e;
    }
    red[tid] = sum; __syncthreads();
    for (int s = 128; s > 0; s >>= 1) {
        if (tid < s) red[tid] += red[tid + s];
        __syncthreads();
    }
    const float inv = 1.0f / red[0];
    for (int c = tid; c < ncols; c += 256) Sr[c] *= inv;
}

// ---------------------------------------------------------------------------
// Reference's transpose(0,2,1).reshape scramble of out_sa (512x128):
//   tmp[f] = out_sa[(f%512)][(f/512)]
// ---------------------------------------------------------------------------
__global__ __launch_bounds__(256)
void scramble_kernel(const float* __restrict__ in, float* __restrict__ out)
{
    const int f = blockIdx.x * blockDim.x + threadIdx.x;
    if (f < 512 * 128) out[f] = in[((f & 511) << 7) + (f >> 9)];
}

__global__ __launch_bounds__(256)
void cat_kernel(const float* __restrict__ sa, const float* __restrict__ da,
                float* __restrict__ cat)
{
    const int i = blockIdx.x * blockDim.x + threadIdx.x;   // n*256 + c
    if (i < 512 * 256) {
        const int n = i >> 8, c = i & 255;
        cat[i] = (c < 128) ? sa[(n << 7) + c] : da[(n << 7) + (c - 128)];
    }
}

// fused = sigmoid(gs)*sa + sigmoid(gd)*da, stored NCHW: out[c*512 + n]
__global__ __launch_bounds__(256)
void fuse_kernel(const float* __restrict__ sa, const float* __restrict__ da,
                 const float* __restrict__ gs, const float* __restrict__ gd,
                 float* __restrict__ out)
{
    const int i = blockIdx.x * blockDim.x + threadIdx.x;   // n*128 + c
    if (i < 512 * 128) {
        const int n = i >> 7, c = i & 127;
        const float fs = 1.0f / (1.0f + __expf(-gs[i]));
        const float fd = 1.0f / (1.0f + __expf(-gd[i]));
        out[(c << 9) + n] = fs * sa[i] + fd * da[i];
    }
}

// ---------------------------------------------------------------------------
extern "C" void kernel_launch(void* const* d_in, const int* in_sizes, int n_in,
                              void* d_out, int out_size, void* d_ws, size_t ws_size,
                              hipStream_t stream)
{
    (void)in_sizes; (void)n_in; (void)out_size; (void)ws_size;

    const float* erp      = (const float*)d_in[0];   // (1,128,16,32)
    const float* ico      = (const float*)d_in[1];   // (1,1280,128)
    const float* icoc     = (const float*)d_in[2];   // (1,1280,3)
    const float* Wq_sa    = (const float*)d_in[3];   // (128,128)
    const float* Wkv_sa   = (const float*)d_in[4];   // (256,128)
    const float* proj_sa  = (const float*)d_in[5];   // (128,128)
    const float* pb_sa    = (const float*)d_in[6];   // (128,)
    const float* Wq_da    = (const float*)d_in[7];
    const float* Wkv_da   = (const float*)d_in[8];
    const float* Wdelta   = (const float*)d_in[9];   // (128,3)
    const float* proj_da  = (const float*)d_in[10];
    const float* pb_da    = (const float*)d_in[11];
    const float* gate_sa  = (const float*)d_in[12];  // (128,256)
    const float* gate_da  = (const float*)d_in[13];
    float* out = (float*)d_out;

    // workspace layout (floats); ~8.4 MB total, all L2-resident
    float* w = (float*)d_ws;
    float* q     = w;                 // 512*128
    float* qsa   = q     + 65536;     // 512*128
    float* qda   = qsa   + 65536;     // 512*128
    float* kvsa  = qda   + 65536;     // 1280*256
    float* kvda  = kvsa  + 327680;    // 1280*256
    float* S     = kvda  + 327680;    // 512*1280 (SA scores, then DA affinity)
    float* outsa = S     + 655360;    // 512*128
    float* outda = outsa + 65536;     // 512*128
    float* tmp   = outda + 65536;     // 512*128 (scrambled out_sa)
    float* sa    = tmp   + 65536;     // 512*128
    float* da    = sa    + 65536;     // 512*128
    float* cat   = da    + 65536;     // 512*256
    float* gs    = cat   + 131072;    // 512*128
    float* gd    = gs    + 65536;     // 512*128
    float* qc    = gd    + 65536;     // 512*3
    float* wsum  = qc    + 1536;      // 3

    const float inv_sqrtC = 0.08838834764831845f;   // 128^-0.5

    // 1) transpose erp -> q, ERP coords, Wdelta column sums
    prep_kernel<<<256, 256, 0, stream>>>(erp, Wdelta, q, qc, wsum);

    // 2) projections (all fp32 WMMA, 16x64 per wave)
    wmma_gemm_f32<<<dim3(2, 32), 32, 0, stream>>>(q,   Wq_sa,  qsa,  nullptr,
        128, 128, 128, 128, 0, 1, 1.0f);
    wmma_gemm_f32<<<dim3(2, 32), 32, 0, stream>>>(q,   Wq_da,  qda,  nullptr,
        128, 128, 128, 128, 0, 1, 1.0f);
    wmma_gemm_f32<<<dim3(4, 80), 32, 0, stream>>>(ico, Wkv_sa, kvsa, nullptr,
        128, 128, 128, 256, 0, 1, 1.0f);
    wmma_gemm_f32<<<dim3(4, 80), 32, 0, stream>>>(ico, Wkv_da, kvda, nullptr,
        128, 128, 128, 256, 0, 1, 1.0f);

    // 3) SA attention: S = (qsa @ ksa^T) * C^-0.5 ; softmax ; outsa = P @ vsa
    wmma_gemm_f32<<<dim3(20, 32), 32, 0, stream>>>(qsa, kvsa, S, nullptr,
        128, 128, 256, 1280, 0, 1, inv_sqrtC);            // ksa = kv cols 0..127
    softmax_rows_kernel<<<512, 256, 0, stream>>>(S, 1280);
    wmma_gemm_f32<<<dim3(2, 32), 32, 0, stream>>>(S, kvsa, outsa, nullptr,
        1280, 1280, 256, 128, 128, 0, 1.0f);              // vsa = kv cols 128..255

    // 4) DA attention: affinity (exp hot loop, async LDS tiles) ; softmax ; PV
    da_affinity_kernel<<<dim3(40, 32), 512, 0, stream>>>(qda, kvda, qc, icoc, wsum, S);
    softmax_rows_kernel<<<512, 256, 0, stream>>>(S, 1280);
    wmma_gemm_f32<<<dim3(2, 32), 32, 0, stream>>>(S, kvda, outda, nullptr,
        1280, 1280, 256, 128, 128, 0, 1.0f);

    // 5) reference's transpose/reshape scramble, then output projections (+bias)
    scramble_kernel<<<256, 256, 0, stream>>>(outsa, tmp);
    wmma_gemm_f32<<<dim3(2, 32), 32, 0, stream>>>(tmp,   proj_sa, sa, pb_sa,
        128, 128, 128, 128, 0, 1, 1.0f);
    wmma_gemm_f32<<<dim3(2, 32), 32, 0, stream>>>(outda, proj_da, da, pb_da,
        128, 128, 128, 128, 0, 1, 1.0f);

    // 6) gates and fusion
    cat_kernel<<<512, 256, 0, stream>>>(sa, da, cat);
    wmma_gemm_f32<<<dim3(2, 32), 32, 0, stream>>>(cat, gate_sa, gs, nullptr,
        256, 256, 256, 128, 0, 1, 1.0f);
    wmma_gemm_f32<<<dim3(2, 32), 32, 0, stream>>>(cat, gate_da, gd, nullptr,
        256, 256, 256, 128, 0, 1, 1.0f);
    fuse_kernel<<<256, 256, 0, stream>>>(sa, da, gs, gd, out);
}